// DGMC_48816598286821
// MI455X (gfx1250) — compile-verified
//
#include <hip/hip_runtime.h>
#include <hip/hip_bf16.h>

typedef float v2f __attribute__((ext_vector_type(2)));
typedef float v8f __attribute__((ext_vector_type(8)));

// ---------------------------------------------------------------------------
// f32 WMMA GEMM: C[M,N] = A[M,K] @ B  (B is [K,N] if !TB, else B is [N,K] and
// we compute A @ B^T). One wave per 16x32 output strip, two accumulators.
// VGPR layouts per CDNA5 ISA 7.12.2 (16x16x4 f32 shape).
// ---------------------------------------------------------------------------
template <bool TB>
__global__ void gemm_f32_wmma(const float* __restrict__ A,
                              const float* __restrict__ B,
                              float* __restrict__ C,
                              int K, int lda, int ldb, int ldc) {
  const int lane = threadIdx.x;        // 0..31 (wave32)
  const int hi   = lane >> 4;          // half-wave select
  const int lo   = lane & 15;
  const int m0   = blockIdx.y * 16;    // 16 rows
  const int n0   = blockIdx.x * 32;    // 32 cols (two 16-wide tiles)

  v8f acc0 = {};
  v8f acc1 = {};

  const float* Arow = A + (size_t)(m0 + lo) * lda;

  for (int k = 0; k < K; k += 4) {
    const int ka = k + 2 * hi;
    v2f a;
    a.x = Arow[ka + 0];
    a.y = Arow[ka + 1];

    v2f b0, b1;
    if (TB) {
      // logical B[k][n] = Bt[n][k], Bt row-major with stride ldb
      const float* Br0 = B + (size_t)(n0 + lo) * ldb + ka;
      const float* Br1 = B + (size_t)(n0 + 16 + lo) * ldb + ka;
      b0.x = Br0[0]; b0.y = Br0[1];
      b1.x = Br1[0]; b1.y = Br1[1];
    } else {
      const float* Bp0 = B + (size_t)ka * ldb + n0 + lo;
      const float* Bp1 = Bp0 + 16;
      b0.x = Bp0[0]; b0.y = Bp0[ldb];
      b1.x = Bp1[0]; b1.y = Bp1[ldb];
    }

    acc0 = __builtin_amdgcn_wmma_f32_16x16x4_f32(false, a, false, b0,
                                                 (short)0, acc0, false, false);
    acc1 = __builtin_amdgcn_wmma_f32_16x16x4_f32(false, a, false, b1,
                                                 (short)0, acc1, false, false);
  }

  // C/D layout: VGPR r holds row m0 + r + 8*hi, col n = lane&15
  float* Cp0 = C + (size_t)(m0 + 8 * hi) * ldc + n0 + lo;
  float* Cp1 = Cp0 + 16;
#pragma unroll
  for (int r = 0; r < 8; ++r) {
    Cp0[(size_t)r * ldc] = acc0[r];
    Cp1[(size_t)r * ldc] = acc1[r];
  }
}

// ---------------------------------------------------------------------------
// Gather rows: out[i][:] = inp[idx[i]][:]
// ---------------------------------------------------------------------------
__global__ void gather_rows(const float* __restrict__ inp,
                            const int* __restrict__ idx,
                            float* __restrict__ out, int C) {
  const int i = blockIdx.x;
  const float* src = inp + (size_t)idx[i] * C;
  for (int c = threadIdx.x; c < C; c += blockDim.x)
    out[(size_t)i * C + c] = src[c];
}

// ---------------------------------------------------------------------------
// Row softmax stats (max, 1/sumexp); optionally writes normalized rows.
// One block per row.
// ---------------------------------------------------------------------------
__global__ void softmax_rows(const float* __restrict__ S,
                             float* __restrict__ rowmax,
                             float* __restrict__ rowinv,
                             float* __restrict__ out, int N) {
  __shared__ float red[256];
  const int row = blockIdx.x;
  const int t = threadIdx.x;
  const float* Sr = S + (size_t)row * N;

  float m = -3.402823466e38f;
  for (int j = t; j < N; j += 256) m = fmaxf(m, Sr[j]);
  red[t] = m;
  __syncthreads();
  for (int s = 128; s > 0; s >>= 1) {
    if (t < s) red[t] = fmaxf(red[t], red[t + s]);
    __syncthreads();
  }
  m = red[0];
  __syncthreads();

  float sum = 0.f;
  for (int j = t; j < N; j += 256) sum += expf(Sr[j] - m);
  red[t] = sum;
  __syncthreads();
  for (int s = 128; s > 0; s >>= 1) {
    if (t < s) red[t] += red[t + s];
    __syncthreads();
  }
  const float inv = 1.0f / red[0];
  if (t == 0) { rowmax[row] = m; rowinv[row] = inv; }
  if (out) {
    for (int j = t; j < N; j += 256)
      out[(size_t)row * N + j] = expf(Sr[j] - m) * inv;
  }
}

// ---------------------------------------------------------------------------
// r_t[j][c] = sum_i softmax(S_hat)[i][j] * r_s[i][c]   (R == 32)
// 256 threads: 8 j-values x 32 channels; S_hat element broadcast per wave.
// ---------------------------------------------------------------------------
__global__ void compute_rt(const float* __restrict__ S_hat,
                           const float* __restrict__ rowmax,
                           const float* __restrict__ rowinv,
                           const float* __restrict__ r_s,
                           float* __restrict__ r_t, int Ns, int Nt) {
  const int c = threadIdx.x & 31;
  const int jl = threadIdx.x >> 5;
  const int j = blockIdx.x * 8 + jl;
  float acc = 0.f;
  for (int i = 0; i < Ns; ++i) {
    const float p = expf(S_hat[(size_t)i * Nt + j] - rowmax[i]) * rowinv[i];
    acc = fmaf(p, r_s[i * 32 + c], acc);
  }
  r_t[j * 32 + c] = acc;
}

__global__ void fill_zero(float* __restrict__ p, int n) {
  const int i = blockIdx.x * blockDim.x + threadIdx.x;
  if (i < n) p[i] = 0.f;
}

// scatter-add over edges: agg[dst[e]][c] += r[src[e]][c]   (R == 32)
__global__ void scatter_add(const int* __restrict__ src,
                            const int* __restrict__ dst,
                            const float* __restrict__ r,
                            float* __restrict__ agg, int E) {
  const int idx = blockIdx.x * blockDim.x + threadIdx.x;
  if (idx >= E * 32) return;
  const int e = idx >> 5;
  const int c = idx & 31;
  atomicAdd(&agg[dst[e] * 32 + c], r[src[e] * 32 + c]);
}

// Y[i][c] = act(bias?[c] + sum_k (X[i][k] + X2?[i][k]) * W[k][c]),  32x32 W
__global__ void linear32(const float* __restrict__ X,
                         const float* __restrict__ X2,
                         const float* __restrict__ W,
                         const float* __restrict__ bias,
                         float* __restrict__ Y, int Nrows, int doRelu) {
  const int idx = blockIdx.x * blockDim.x + threadIdx.x;
  if (idx >= Nrows * 32) return;
  const int i = idx >> 5;
  const int c = idx & 31;
  float acc = bias ? bias[c] : 0.f;
#pragma unroll 8
  for (int k = 0; k < 32; ++k) {
    float x = X[i * 32 + k];
    if (X2) x += X2[i * 32 + k];
    acc = fmaf(x, W[k * 32 + c], acc);
  }
  Y[idx] = doRelu ? fmaxf(acc, 0.f) : acc;
}

// S_hat[i][j] += bm2 + sum_c relu(A[i][c] - B[j][c]) * Wm2[c]
__global__ void delta_update(const float* __restrict__ A,
                             const float* __restrict__ B,
                             const float* __restrict__ Wm2,
                             const float* __restrict__ bm2,
                             float* __restrict__ S_hat, int Nt) {
  __shared__ float sA[16 * 32];
  __shared__ float sB[16 * 32];
  __shared__ float sW[32];
  const int t = threadIdx.x;            // 256 threads
  const int i0 = blockIdx.y * 16;
  const int j0 = blockIdx.x * 16;
  sA[t] = A[i0 * 32 + t];
  sA[t + 256] = A[i0 * 32 + t + 256];
  sB[t] = B[j0 * 32 + t];
  sB[t + 256] = B[j0 * 32 + t + 256];
  if (t < 32) sW[t] = Wm2[t];
  __syncthreads();
  const int ti = t >> 4;
  const int tj = t & 15;
  float acc = bm2[0];
#pragma unroll
  for (int c = 0; c < 32; ++c) {
    const float h = sA[ti * 32 + c] - sB[tj * 32 + c];
    acc = fmaf(fmaxf(h, 0.f), sW[c], acc);
  }
  S_hat[(size_t)(i0 + ti) * Nt + (j0 + tj)] += acc;
}

// ---------------------------------------------------------------------------
extern "C" void kernel_launch(void* const* d_in, const int* in_sizes, int n_in,
                              void* d_out, int out_size, void* d_ws,
                              size_t ws_size, hipStream_t stream) {
  const float* inputs   = (const float*)d_in[0];
  const int*   index_n1 = (const int*)d_in[1];
  const int*   index_n2 = (const int*)d_in[2];
  const int*   ei_s     = (const int*)d_in[3];
  const int*   ei_t     = (const int*)d_in[4];
  const float* W1       = (const float*)d_in[5];
  const float* W2       = (const float*)d_in[6];
  const float* W3       = (const float*)d_in[7];
  const float* b3       = (const float*)d_in[8];
  const float* Wm1      = (const float*)d_in[9];
  const float* bm1      = (const float*)d_in[10];
  const float* Wm2      = (const float*)d_in[11];
  const float* bm2      = (const float*)d_in[12];
  const float* rs_all   = (const float*)d_in[13];

  const int Ns = 1024, Nt = 1024, Cin = 128, C1 = 256, Ch = 512, R = 32;
  const int E = in_sizes[3] / 2;
  const int NUM_STEPS = 2;

  float* ws = (float*)d_ws;
  float* Xs     = ws; ws += Ns * Cin;
  float* Xt     = ws; ws += Nt * Cin;
  float* hs     = ws; ws += Ns * Ch;
  float* ht     = ws; ws += Nt * Ch;
  float* Shat   = ws; ws += (size_t)Ns * Nt;
  float* rowmax = ws; ws += Ns;
  float* rowinv = ws; ws += Ns;
  float* rt     = ws; ws += Nt * R;
  float* aggs   = ws; ws += Ns * R;
  float* aggt   = ws; ws += Nt * R;
  float* os_    = ws; ws += Ns * R;
  float* ot_    = ws; ws += Nt * R;
  float* As     = ws; ws += Ns * R;
  float* Bt     = ws; ws += Nt * R;

  float* S0 = (float*)d_out;
  float* SL = (float*)d_out + (size_t)Ns * Nt;

  // 1) gather entity rows
  gather_rows<<<Ns, 128, 0, stream>>>(inputs, index_n1, Xs, Cin);
  gather_rows<<<Nt, 128, 0, stream>>>(inputs, index_n2, Xt, Cin);

  // 2) h = [X@W1 | X@W2]  (WMMA f32, ldc=512 interleaves the concat)
  dim3 g1(C1 / 32, Ns / 16);
  gemm_f32_wmma<false><<<g1, 32, 0, stream>>>(Xs, W1, hs,       Cin, Cin, C1, Ch);
  gemm_f32_wmma<false><<<g1, 32, 0, stream>>>(Xs, W2, hs + C1,  Cin, Cin, C1, Ch);
  gemm_f32_wmma<false><<<g1, 32, 0, stream>>>(Xt, W1, ht,       Cin, Cin, C1, Ch);
  gemm_f32_wmma<false><<<g1, 32, 0, stream>>>(Xt, W2, ht + C1,  Cin, Cin, C1, Ch);

  // 3) S_hat = h_s @ h_t^T  (WMMA f32, transposed-B path)
  dim3 g2(Nt / 32, Ns / 16);
  gemm_f32_wmma<true><<<g2, 32, 0, stream>>>(hs, ht, Shat, Ch, Ch, Ch, Nt);

  // 4) S_0 = softmax(S_hat) (stats reused for step-0 r_t)
  softmax_rows<<<Ns, 256, 0, stream>>>(Shat, rowmax, rowinv, S0, Nt);

  // 5) consensus steps
  for (int step = 0; step < NUM_STEPS; ++step) {
    if (step > 0)
      softmax_rows<<<Ns, 256, 0, stream>>>(Shat, rowmax, rowinv, nullptr, Nt);
    const float* rs = rs_all + (size_t)step * Ns * R;

    compute_rt<<<Nt / 8, 256, 0, stream>>>(Shat, rowmax, rowinv, rs, rt, Ns, Nt);

    fill_zero<<<(Ns * R) / 256, 256, 0, stream>>>(aggs, Ns * R);
    fill_zero<<<(Nt * R) / 256, 256, 0, stream>>>(aggt, Nt * R);
    scatter_add<<<(E * R + 255) / 256, 256, 0, stream>>>(ei_s, ei_s + E, rs, aggs, E);
    scatter_add<<<(E * R + 255) / 256, 256, 0, stream>>>(ei_t, ei_t + E, rt, aggt, E);

    // o = relu((r + agg) @ W3 + b3)
    linear32<<<(Ns * R) / 256, 256, 0, stream>>>(rs, aggs, W3, b3, os_, Ns, 1);
    linear32<<<(Nt * R) / 256, 256, 0, stream>>>(rt, aggt, W3, b3, ot_, Nt, 1);

    // separable MLP precompute: A = o_s@Wm1 + bm1 ; B = o_t@Wm1
    linear32<<<(Ns * R) / 256, 256, 0, stream>>>(os_, nullptr, Wm1, bm1, As, Ns, 0);
    linear32<<<(Nt * R) / 256, 256, 0, stream>>>(ot_, nullptr, Wm1, nullptr, Bt, Nt, 0);

    delta_update<<<dim3(Nt / 16, Ns / 16), 256, 0, stream>>>(As, Bt, Wm2, bm2,
                                                             Shat, Nt);
  }

  // 6) S_L = softmax(S_hat)
  softmax_rows<<<Ns, 256, 0, stream>>>(Shat, rowmax, rowinv, SL, Nt);
}